// SimplePolynomial_58548994179107
// MI455X (gfx1250) — compile-verified
//
#include <hip/hip_runtime.h>

// out = (x^2) * 0.1 over 67.1M fp32 elements.
// Pure streaming: 537 MB HBM traffic, 0.25 FLOP/byte -> ~23us floor at
// 23.3 TB/s. Strategy: b128 non-temporal loads/stores (stream > 192MB L2,
// zero reuse), 4 loads in flight per iteration, coalesced strided
// grid-stride loop, 32-bit indexing (byte extent 268MB < 2^31) so addresses
// use the SADDR + 32-bit-voffset form instead of 64-bit VALU adds.

typedef __attribute__((ext_vector_type(4))) float v4f;

#ifndef POLY_SCALAR
#define POLY_SCALAR 0.1f
#endif

__global__ __launch_bounds__(256) void
SimplePolynomial_58548994179107_kernel(const float* __restrict__ xin,
                                       float* __restrict__ xout,
                                       int n)
{
    const unsigned tid      = blockIdx.x * blockDim.x + threadIdx.x;
    const unsigned nthreads = gridDim.x * blockDim.x;
    const unsigned n4       = (unsigned)n >> 2;   // float4 chunks

    const v4f* __restrict__ x = (const v4f*)xin;
    v4f* __restrict__ o       = (v4f*)xout;

    unsigned i = tid;

    // Main loop: 4 independent global_load_b128 (TH_NT) in flight before any
    // compute/store. Each wave moves 2KB of loads + 2KB of stores per
    // iteration; lanes stay contiguous so every vmem op is a full 512B burst.
    for (; i + 3u * nthreads < n4; i += 4u * nthreads) {
        v4f a0 = __builtin_nontemporal_load(&x[i]);
        v4f a1 = __builtin_nontemporal_load(&x[i +      nthreads]);
        v4f a2 = __builtin_nontemporal_load(&x[i + 2u * nthreads]);
        v4f a3 = __builtin_nontemporal_load(&x[i + 3u * nthreads]);

        a0 = (a0 * a0) * POLY_SCALAR;
        a1 = (a1 * a1) * POLY_SCALAR;
        a2 = (a2 * a2) * POLY_SCALAR;
        a3 = (a3 * a3) * POLY_SCALAR;

        __builtin_nontemporal_store(a0, &o[i]);
        __builtin_nontemporal_store(a1, &o[i +      nthreads]);
        __builtin_nontemporal_store(a2, &o[i + 2u * nthreads]);
        __builtin_nontemporal_store(a3, &o[i + 3u * nthreads]);
    }

    // Remainder float4 chunks
    for (; i < n4; i += nthreads) {
        v4f a = __builtin_nontemporal_load(&x[i]);
        a = (a * a) * POLY_SCALAR;
        __builtin_nontemporal_store(a, &o[i]);
    }

    // Scalar tail: n % 4 elements (0 for the 64M-element reference shape).
    const unsigned base = n4 << 2;
    if (base + tid < (unsigned)n) {
        float v = xin[base + tid];
        xout[base + tid] = (v * v) * POLY_SCALAR;
    }
}

extern "C" void kernel_launch(void* const* d_in, const int* in_sizes, int n_in,
                              void* d_out, int out_size, void* d_ws, size_t ws_size,
                              hipStream_t stream)
{
    (void)n_in; (void)d_ws; (void)ws_size; (void)out_size;

    const float* x   = (const float*)d_in[0];
    float*       out = (float*)d_out;
    const int    n   = in_sizes[0];

    const int threads = 256;                  // 8 wave32 per block
    int blocks = 8192;                        // 64Ki waves: saturates HBM
    const long long n4     = (long long)((unsigned)n >> 2);
    const long long needed = (n4 + threads - 1) / threads;
    if (needed < (long long)blocks) blocks = (int)(needed > 0 ? needed : 1);

    SimplePolynomial_58548994179107_kernel<<<blocks, threads, 0, stream>>>(x, out, n);
}